// CM_Module_1194000908895
// MI455X (gfx1250) — compile-verified
//
#include <hip/hip_runtime.h>
#include <hip/hip_bf16.h>

// ---------------------------------------------------------------------------
// CM_Module: masked cross-frame matching on MI455X (gfx1250, wave32).
//
// Pipeline (all f32):
//   K1 resize_mask : antialiased 4x bilinear downsample + >0.5 threshold
//   K2 gs_partial  : per-(b, pixel-block, c-segment) partials of
//                    gs[b,t] = sum_{c,hw} tv*rv * t_feat * r_feats and v_sum.
//                    Channel-dot done on the matrix pipe:
//                    dot tiles = diag( t_featT(16px x 4c)  x  r_feats(4c x 16px) )
//                    via V_WMMA_F32_16X16X4_F32, accumulator diagonal extracted
//                    per the CDNA5 16x16 f32 C/D VGPR layout.
//   K3 gs_finalize : deterministic tree-free reduction of partials + EPS logic
//   K4 combine     : per-pixel masked softmax over t, c_out, c_mask, concat.
// ---------------------------------------------------------------------------

typedef __attribute__((ext_vector_type(2))) float v2f;
typedef __attribute__((ext_vector_type(8))) float v8f;

#define EPS_K   1e-4f
#define Bsz     8
#define Csz     128
#define Tsz     8
#define HW      4096           // 64*64
#define CSTRIDE (9 * 4096)     // Tt*H*W stride between channels in `values`

// ------------------------- K1: resize + threshold --------------------------
// jax.image.resize(bilinear, antialias) 256->64: triangle kernel, radius 4,
// 8 taps/dim, edge-renormalized. One thread per output mask element.
__global__ __launch_bounds__(256) void resize_mask_kernel(
    const float* __restrict__ tvmap, const float* __restrict__ rvmaps,
    float* __restrict__ tvmask, float* __restrict__ rvmask)
{
  int id = blockIdx.x * 256 + threadIdx.x;          // exactly B*9*4096 threads
  int p = id & 4095;
  int slot = id >> 12;
  int b = slot / 9, m = slot % 9;
  int oy = p >> 6, ox = p & 63;
  const float* src = (m == 0) ? (tvmap + (size_t)b * 65536)
                              : (rvmaps + ((size_t)b * 8 + (m - 1)) * 65536);
  float cy = 4.f * oy + 1.5f, cx = 4.f * ox + 1.5f;
  float val = 0.f, wsy = 0.f;
  for (int jy = 4 * oy - 2; jy <= 4 * oy + 5; ++jy) {
    if (jy < 0 || jy > 255) continue;
    float wy = 1.f - fabsf((float)jy - cy) * 0.25f;
    wsy += wy;
    float rowv = 0.f, wsx = 0.f;
    for (int jx = 4 * ox - 2; jx <= 4 * ox + 5; ++jx) {
      if (jx < 0 || jx > 255) continue;
      float wx = 1.f - fabsf((float)jx - cx) * 0.25f;
      wsx += wx;
      rowv += wx * src[jy * 256 + jx];
    }
    val += wy * (rowv / wsx);
  }
  val /= wsy;
  float mk = (val > 0.5f) ? 1.f : 0.f;
  if (m == 0) tvmask[(size_t)b * HW + p] = mk;
  else        rvmask[((size_t)b * 8 + (m - 1)) * HW + p] = mk;
}

// --------------------- K2: gs/v_sum partials (WMMA) ------------------------
// grid = (B*32, 4): 32 pixel-blocks of 128 px per b, 4 channel segments of 32.
// 8 waves/block, one 16-pixel tile per wave.
// WMMA layout used (CDNA5 ISA 7.12.2, 32-bit 16x4 A / 4x16 B / 16x16 C,D):
//   A: lane(0-15)=M row, VGPR0/1 = K 0/1 (lanes 0-15) or K 2/3 (lanes 16-31)
//   B: lane(0-15)=N col, same K striping
//   D: VGPR r, lanes 0-15 -> (M=r, N=lane); lanes 16-31 -> (M=8+r, N=lane-16)
//   => diag[n]: n<8 at (lane=n, reg=n); n>=8 at (lane=n+16, reg=n-8)
__global__ __launch_bounds__(256) void gs_partial_kernel(
    const float* __restrict__ values, const float* __restrict__ tvmask,
    const float* __restrict__ rvmask, float* __restrict__ gspart,
    float* __restrict__ vspart)
{
  __shared__ float red[8][8][2];                   // [wave][t][{gs, vsum}]
  int b    = blockIdx.x >> 5;
  int pblk = blockIdx.x & 31;
  int cseg = blockIdx.y;
  int wave = threadIdx.x >> 5;
  int lane = threadIdx.x & 31;
  int n    = lane & 15;                            // pixel-in-tile (M and N)
  int p    = pblk * 128 + wave * 16 + n;           // pixel in [0,4096)
  int koff = (lane < 16) ? 0 : 2;

  const float* tf = values + (size_t)b * Csz * CSTRIDE + p;   // + c*CSTRIDE
  const float* rf = tf + HW;                                  // tt = 1 + t

  v8f z = {0.f, 0.f, 0.f, 0.f, 0.f, 0.f, 0.f, 0.f};
  v8f acc[8] = {z, z, z, z, z, z, z, z};

  int clo = cseg * 32, chi = clo + 32;
  for (int c0 = clo; c0 < chi; c0 += 4) {
    const float* ta = tf + (size_t)(c0 + koff) * CSTRIDE;
    v2f a;
    a.x = ta[0];
    a.y = ta[CSTRIDE];                             // channel c0+koff+1
    const float* ra = rf + (size_t)(c0 + koff) * CSTRIDE;
#pragma unroll
    for (int t = 0; t < 8; ++t) {                  // A-tile reused for all t
      v2f bm;
      bm.x = ra[t * HW];
      bm.y = ra[CSTRIDE + t * HW];
      acc[t] = __builtin_amdgcn_wmma_f32_16x16x4_f32(
          false, a, false, bm, (short)0, acc[t], false, false);
    }
  }

  float tvv = tvmask[(size_t)b * HW + p];
  int r = lane & 7;
  bool diagLane = (lane < 8) || (lane >= 24);      // lanes holding diag(D)
#pragma unroll
  for (int t = 0; t < 8; ++t) {
    float rvv = rvmask[((size_t)b * 8 + t) * HW + p];
    float vm = tvv * rvv;
    v8f A = acc[t];
    float d = A[0];
    d = (r == 1) ? A[1] : d;
    d = (r == 2) ? A[2] : d;
    d = (r == 3) ? A[3] : d;
    d = (r == 4) ? A[4] : d;
    d = (r == 5) ? A[5] : d;
    d = (r == 6) ? A[6] : d;
    d = (r == 7) ? A[7] : d;
    float dv  = diagLane ? d * vm : 0.f;           // gs contribution
    float vsv = (lane < 16) ? vm : 0.f;            // v_sum (count pixels once)
#pragma unroll
    for (int s = 16; s > 0; s >>= 1) {
      dv  += __shfl_xor(dv, s, 32);
      vsv += __shfl_xor(vsv, s, 32);
    }
    if (lane == 0) { red[wave][t][0] = dv; red[wave][t][1] = vsv; }
  }
  __syncthreads();
  if (threadIdx.x < 8) {
    int t = threadIdx.x;
    float g = 0.f, v = 0.f;
#pragma unroll
    for (int w = 0; w < 8; ++w) { g += red[w][t][0]; v += red[w][t][1]; }
    gspart[((size_t)b * 8 + t) * 128 + cseg * 32 + pblk] = g;
    if (cseg == 0) vspart[((size_t)b * 8 + t) * 32 + pblk] = v;
  }
}

// ----------------------- K3: finalize gs (tiny) ----------------------------
__global__ void gs_finalize_kernel(const float* __restrict__ gspart,
                                   const float* __restrict__ vspart,
                                   float* __restrict__ gsfin)
{
  int i = threadIdx.x;                             // 64 threads = B*T
  float g = 0.f;
  for (int k = 0; k < 128; ++k) g += gspart[i * 128 + k];
  float v = 0.f;
  for (int k = 0; k < 32; ++k) v += vspart[i * 32 + k];
  float zz = (v < EPS_K) ? 1.f : 0.f;
  g = (zz > 0.f) ? 0.f : g;
  gsfin[i] = g / (v + zz) / (float)Csz;
}

// ------------- K4: softmax over t, c_out, c_mask, concat -------------------
__global__ __launch_bounds__(256) void combine_kernel(
    const float* __restrict__ values, const float* __restrict__ rvmask,
    const float* __restrict__ gsfin, float* __restrict__ out)
{
  int idx = blockIdx.x * 256 + threadIdx.x;        // B*4096 threads
  int b = idx >> 12;
  int p = idx & 4095;

  float rvm[8], me[8];
  float mx = -3.402823466e38f;
#pragma unroll
  for (int t = 0; t < 8; ++t) {
    rvm[t] = rvmask[((size_t)b * 8 + t) * HW + p];
    float mv = gsfin[b * 8 + t] * rvm[t];
    me[t] = mv;
    mx = fmaxf(mx, mv);
  }
  float s = 0.f;
#pragma unroll
  for (int t = 0; t < 8; ++t) {
    me[t] = expf(me[t] - mx) * rvm[t];
    s += me[t];
  }
  s += (s < EPS_K) ? 1.f : 0.f;
  float inv = 1.f / s;
  float cmsum = 0.f;
#pragma unroll
  for (int t = 0; t < 8; ++t) {
    me[t] *= inv;                                  // c_match
    cmsum += me[t] * rvm[t];
  }
  float cmask = 1.f - cmsum;

  const float* base = values + (size_t)b * Csz * CSTRIDE + p;
  float* ob = out + (size_t)b * (2 * Csz + 1) * HW + p;
  for (int c = 0; c < Csz; ++c) {
    const float* vc = base + (size_t)c * CSTRIDE;
    float tfv = vc[0];
    float co = 0.f;
#pragma unroll
    for (int t = 0; t < 8; ++t) co += vc[(t + 1) * HW] * me[t];
    ob[(size_t)c * HW] = tfv;                      // t_feat copy
    ob[(size_t)(Csz + c) * HW] = co;               // c_out
  }
  ob[(size_t)(2 * Csz) * HW] = cmask;              // c_mask channel
  out[(size_t)Bsz * (2 * Csz + 1) * HW + (size_t)b * HW + p] = cmask; // 2nd out
}

// ---------------------------------------------------------------------------
extern "C" void kernel_launch(void* const* d_in, const int* in_sizes, int n_in,
                              void* d_out, int out_size, void* d_ws, size_t ws_size,
                              hipStream_t stream) {
  const float* values = (const float*)d_in[0];   // (8,128,9,64,64)
  const float* tvmap  = (const float*)d_in[1];   // (8,1,256,256)
  const float* rvmaps = (const float*)d_in[2];   // (8,1,8,256,256)
  float* out = (float*)d_out;                    // (8,257,64,64) ++ (8,1,64,64)

  // workspace layout (~1.2 MB of f32)
  float* ws     = (float*)d_ws;
  float* tvmask = ws;                            // B*HW          = 32768
  float* rvmask = tvmask + Bsz * HW;             // B*T*HW        = 262144
  float* gspart = rvmask + Bsz * Tsz * HW;       // B*T*128       = 8192
  float* vspart = gspart + Bsz * Tsz * 128;      // B*T*32        = 2048
  float* gsfin  = vspart + Bsz * Tsz * 32;       // B*T           = 64

  resize_mask_kernel<<<(Bsz * 9 * HW) / 256, 256, 0, stream>>>(
      tvmap, rvmaps, tvmask, rvmask);

  dim3 g2(Bsz * 32, 4);
  gs_partial_kernel<<<g2, 256, 0, stream>>>(values, tvmask, rvmask,
                                            gspart, vspart);

  gs_finalize_kernel<<<1, 64, 0, stream>>>(gspart, vspart, gsfin);

  combine_kernel<<<(Bsz * HW) / 256, 256, 0, stream>>>(values, rvmask,
                                                       gsfin, out);
}